// TopKLayer_54382875902679
// MI455X (gfx1250) — compile-verified
//
#include <hip/hip_runtime.h>
#include <stdint.h>

// Top-k (k = 313 of 3136) magnitude sparsification per (n,c) row.
// One workgroup per row; row staged into LDS via the Tensor Data Mover,
// exact threshold found with a 4-round radix select on abs-bit patterns,
// output written with 128-bit non-temporal stores.

#define HW      3136            // 56*56
#define HW4     (HW / 4)        // 784 float4 chunks
#define K_TOP   313             // max(1, int(0.1 * 3136))
#define NT      256             // 8 waves of 32

typedef __attribute__((ext_vector_type(4))) float        f32x4;
typedef __attribute__((ext_vector_type(4))) unsigned int u32x4;
typedef __attribute__((ext_vector_type(4))) int          i32x4;
typedef __attribute__((ext_vector_type(8))) int          i32x8;

__device__ __forceinline__ unsigned absbits(float f) {
    return __float_as_uint(f) & 0x7fffffffu;
}

__global__ __launch_bounds__(NT) void topk_mask_kernel(const float* __restrict__ x,
                                                       float* __restrict__ out) {
    __shared__ __align__(16) float row[HW];   // 12544 B row buffer
    __shared__ unsigned hist[256];
    __shared__ unsigned sfx[256];
    __shared__ unsigned bc[3];

    const int tid = threadIdx.x;
    const float* src = x   + (size_t)blockIdx.x * HW;
    float*       dst = out + (size_t)blockIdx.x * HW;

    // ---------------- Stage row into LDS via TDM ----------------
#if defined(__gfx1250__) && __has_builtin(__builtin_amdgcn_tensor_load_to_lds) && \
    __has_builtin(__builtin_amdgcn_s_wait_tensorcnt)
    if (tid < 32) {  // one wave issues the DMA (tensor ops ignore EXEC)
        // Low 32 bits of the flat LDS address == LDS byte offset (ISA 10.2).
        unsigned lds_off = (unsigned)(uintptr_t)(&row[0]);
        unsigned long long ga = (unsigned long long)(uintptr_t)src;

        u32x4 g0;
        g0[0] = 1u;                                    // count=1, user descriptor
        g0[1] = lds_off;                               // lds_addr
        g0[2] = (unsigned)(ga & 0xffffffffu);          // global_addr[31:0]
        g0[3] = (unsigned)((ga >> 32) & 0x01ffffffu)   // global_addr[56:32]
              | (2u << 30);                            // type = 2 ("image")

        i32x8 g1;
        g1[0] = (int)(2u << 16);                       // wg_mask=0, data_size=2 (4B)
        g1[1] = (int)((HW & 0xffffu) << 16);           // tensor_dim0[15:0]
        g1[2] = (int)(((HW >> 16) & 0xffffu)           // tensor_dim0[31:16]
              | (1u << 16));                           // tensor_dim1 = 1
        g1[3] = (int)((HW & 0xffffu) << 16);           // tile_dim0 = 3136
        g1[4] = 1;                                     // tile_dim1 = 1, tile_dim2 = 0
        g1[5] = (int)HW;                               // tensor_dim0_stride[31:0]
        g1[6] = 0;                                     // stride0 hi, stride1 lo
        g1[7] = 0;                                     // stride1 hi

        i32x4 gz4 = {0, 0, 0, 0};                      // groups 2/3 unused (<=2D)
        i32x8 gz8 = {0, 0, 0, 0, 0, 0, 0, 0};
        __builtin_amdgcn_tensor_load_to_lds(g0, g1, gz4, gz4, gz8, 0);
        __builtin_amdgcn_s_wait_tensorcnt(0);
    }
#else
    for (int i = tid; i < HW4; i += NT)
        ((f32x4*)row)[i] = ((const f32x4*)src)[i];
#endif
    __syncthreads();

    // ---------------- Radix select: K_TOP-th largest |x| ----------------
    unsigned prefixVal = 0;   // threshold bits fixed so far (top-down)
    unsigned remaining = K_TOP;
    unsigned cntEq     = 0;

    const int shifts[4] = {23, 15, 7, 0};
    const int widths[4] = {8, 8, 8, 7};

    #pragma unroll
    for (int round = 0; round < 4; ++round) {
        const int      shift     = shifts[round];
        const int      width     = widths[round];
        const unsigned binMask   = (1u << width) - 1u;
        const unsigned fixedMask = (round == 0)
            ? 0u : (0x7fffffffu & ~((1u << (shift + width)) - 1u));

        hist[tid] = 0u;
        __syncthreads();

        const f32x4* rv = (const f32x4*)row;
        for (int i = tid; i < HW4; i += NT) {
            f32x4 v = rv[i];
            #pragma unroll
            for (int c = 0; c < 4; ++c) {
                unsigned u = absbits(v[c]);
                if ((u & fixedMask) == prefixVal)
                    atomicAdd(&hist[(u >> shift) & binMask], 1u);
            }
        }
        __syncthreads();

        // Suffix-inclusive scan (descending order) over 256 bins.
        sfx[tid] = hist[tid];
        __syncthreads();
        #pragma unroll
        for (int d = 1; d < 256; d <<= 1) {
            unsigned add = (tid + d < 256) ? sfx[tid + d] : 0u;
            __syncthreads();
            sfx[tid] += add;
            __syncthreads();
        }

        // Exactly one bin satisfies: count-above < remaining <= count-at-or-above.
        {
            unsigned ge    = sfx[tid];
            unsigned above = (tid < 255) ? sfx[tid + 1] : 0u;
            if (above < remaining && ge >= remaining) {
                bc[0] = (unsigned)tid;   // selected bin
                bc[1] = above;           // elements strictly above this bin
                bc[2] = hist[tid];       // candidates inside this bin
            }
        }
        __syncthreads();
        prefixVal |= bc[0] << shift;
        remaining -= bc[1];
        cntEq      = bc[2];
        __syncthreads();
    }

    const unsigned T      = prefixVal;   // exact abs-bits of k-th largest
    const unsigned keepEq = remaining;   // how many T-valued elements survive

    // ---------------- Rare tie resolution (exact duplicates of T) -------------
    if (cntEq != keepEq && T != 0u) {
        unsigned km  = 0u;               // kill mask over this thread's slots
        int      slot = 0;
        for (int i = tid; i < HW; i += NT, ++slot) {
            if (absbits(row[i]) == T) {
                unsigned rank = 0;       // ties at lower index win (matches top_k)
                for (int j = 0; j < i; ++j) rank += (absbits(row[j]) == T);
                if (rank >= keepEq) km |= (1u << slot);
            }
        }
        __syncthreads();
        slot = 0;
        for (int i = tid; i < HW; i += NT, ++slot)
            if (km & (1u << slot)) row[i] = 0.0f;
        __syncthreads();
    }

    // ---------------- Write: keep |x| >= T, else zero; 128-bit NT stores ------
    {
        const f32x4* rv = (const f32x4*)row;
        f32x4*       ov = (f32x4*)dst;
        for (int i = tid; i < HW4; i += NT) {
            f32x4 v = rv[i];
            #pragma unroll
            for (int c = 0; c < 4; ++c)
                v[c] = (absbits(v[c]) >= T) ? v[c] : 0.0f;
            __builtin_nontemporal_store(v, &ov[i]);
        }
    }
}

extern "C" void kernel_launch(void* const* d_in, const int* in_sizes, int n_in,
                              void* d_out, int out_size, void* d_ws, size_t ws_size,
                              hipStream_t stream) {
    (void)n_in; (void)out_size; (void)d_ws; (void)ws_size;
    const float* x = (const float*)d_in[0];
    float* out = (float*)d_out;
    const int rows = in_sizes[0] / HW;   // 32 * 256 = 8192
    topk_mask_kernel<<<rows, NT, 0, stream>>>(x, out);
}